// MolPropModel_20890720928255
// MI455X (gfx1250) — compile-verified
//
#include <hip/hip_runtime.h>
#include <hip/hip_bf16.h>
#include <math.h>

// ---------------- constants matching the reference ----------------
#define HID 128
#define NHEAD 8
#define DH 16
#define WT_STRIDE 132   // padded LDS stride: 132 mod 64 = 4 -> conflict-free ds_load_b64

typedef float v2f __attribute__((ext_vector_type(2)));
typedef float v8f __attribute__((ext_vector_type(8)));

// ---------------- helpers ----------------
__device__ __forceinline__ void atomicMaxFloat(float* addr, float val) {
    // Ordered-equivalent integer trick for IEEE-754 floats.
    if (val >= 0.0f) {
        atomicMax((int*)addr, __float_as_int(val));
    } else {
        atomicMin((unsigned int*)addr, __float_as_uint(val));
    }
}

__device__ __forceinline__ float waveReduceSum(float v) {
    #pragma unroll
    for (int m = 16; m > 0; m >>= 1) v += __shfl_xor(v, m, 32);
    return v;
}
__device__ __forceinline__ float waveReduceMax(float v) {
    #pragma unroll
    for (int m = 16; m > 0; m >>= 1) v = fmaxf(v, __shfl_xor(v, m, 32));
    return v;
}
__device__ __forceinline__ float leaky02(float v) { return v > 0.0f ? v : 0.2f * v; }
__device__ __forceinline__ float eluf(float v)    { return v > 0.0f ? v : expf(v) - 1.0f; }

// ---------------- GEMM via V_WMMA_F32_16X16X4_F32 ----------------
// XL = X * Wl ; XR = X * Wr    (X: [n,128], W: [128,128] row-major)
// Wave pairs: even waves -> Wl, odd waves -> Wr; each wave owns a 16-row tile
// (64 acc VGPRs), leaving room for explicit double-buffered A/B fragments so
// ds_load latency overlaps the matrix pipe instead of serializing on waits.
// Wl/Wr are staged TRANSPOSED in dynamic LDS (~135 KB/WG; 2 WGs per 320 KB WGP)
// so each B-fragment pair (k,k+1) is one conflict-free ds_load_b64.
__global__ __launch_bounds__(256) void k_gemm_dual(
    const float* __restrict__ X, const float* __restrict__ Wl,
    const float* __restrict__ Wr, float* __restrict__ XL,
    float* __restrict__ XR, int nrows)
{
    extern __shared__ float lds[];
    float* WlT = lds;
    float* WrT = lds + HID * WT_STRIDE;

    // stage transposed weights: WT[col*STRIDE + k] = W[k*HID + col]
    for (int i = threadIdx.x; i < HID * HID; i += blockDim.x) {
        const int row = i >> 7;          // k
        const int col = i & (HID - 1);   // n
        WlT[col * WT_STRIDE + row] = Wl[i];
        WrT[col * WT_STRIDE + row] = Wr[i];
    }
    __syncthreads();

    const int wib  = threadIdx.x >> 5;   // wave in block: 0..7
    const int lane = threadIdx.x & 31;
    const int mat  = wib & 1;            // 0 -> Wl, 1 -> Wr
    const int tile = blockIdx.x * 4 + (wib >> 1);
    if (tile * 16 >= nrows) return;      // wave-uniform: EXEC all-ones at WMMAs

    const float* WT  = mat ? WrT : WlT;
    float*       OUT = mat ? XR  : XL;

    const int r    = lane & 15;          // 0..15
    const int half = lane >> 4;          // 0/1 (lane group)
    const int rowA = tile * 16 + r;
    const float* Arow = X + rowA * HID + half * 2;

    v8f acc[8];
    #pragma unroll
    for (int t = 0; t < 8; ++t)
        #pragma unroll
        for (int i = 0; i < 8; ++i) acc[t][i] = 0.0f;

    // double-buffered fragments: A from global, B from LDS
    v2f abuf[2];
    v2f bbuf[2][8];
    abuf[0] = *(const v2f*)(Arow);
    #pragma unroll
    for (int t = 0; t < 8; ++t)
        bbuf[0][t] = *(const v2f*)&WT[(t * 16 + r) * WT_STRIDE + half * 2];

    #pragma unroll
    for (int kk = 0; kk < HID; kk += 4) {
        const int cur = (kk >> 2) & 1;
        const int nxt = cur ^ 1;
        if (kk + 4 < HID) {              // statically elided on the last step
            abuf[nxt] = *(const v2f*)(Arow + kk + 4);
            #pragma unroll
            for (int t = 0; t < 8; ++t)
                bbuf[nxt][t] = *(const v2f*)
                    &WT[(t * 16 + r) * WT_STRIDE + kk + 4 + half * 2];
        }
        #pragma unroll
        for (int t = 0; t < 8; ++t)
            acc[t] = __builtin_amdgcn_wmma_f32_16x16x4_f32(
                false, abuf[cur], false, bbuf[cur][t], (short)0, acc[t],
                false, false);
    }

    // C/D layout: VGPR i -> rows (i + half*8), column = lane&15 within tile n
    #pragma unroll
    for (int t = 0; t < 8; ++t) {
        const int col = t * 16 + r;
        #pragma unroll
        for (int i = 0; i < 8; ++i)
            OUT[(tile * 16 + half * 8 + i) * HID + col] = acc[t][i];
    }
}

// ---------------- per-layer init: zero aggregate, -inf maxes, zero denoms ----
__global__ void k_init_layer(float* __restrict__ xbuf, float* __restrict__ m,
                             float* __restrict__ denom, int n) {
    int i = blockIdx.x * blockDim.x + threadIdx.x;
    if (i < n * HID) xbuf[i] = 0.0f;
    if (i < n * NHEAD) { m[i] = -INFINITY; denom[i] = 0.0f; }
}

// ---------------- edge pass 1: attention logits + segment max -------------
__global__ __launch_bounds__(256) void k_edge_score(
    const float* __restrict__ xl, const float* __restrict__ xr,
    const int* __restrict__ src, const int* __restrict__ dst,
    const float* __restrict__ att, float* __restrict__ esc,
    float* __restrict__ m, int EH)
{
    int t = blockIdx.x * blockDim.x + threadIdx.x;
    if (t >= EH) return;
    const int e = t >> 3, h = t & 7;
    const int s = src[e], d = dst[e];
    const float4* pl = (const float4*)(xl + s * HID + h * DH);
    const float4* pr = (const float4*)(xr + d * HID + h * DH);
    const float4* pa = (const float4*)(att + h * DH);
    float acc = 0.0f;
    #pragma unroll
    for (int q = 0; q < 4; ++q) {
        float4 a = pl[q], b = pr[q], w = pa[q];
        acc += leaky02(a.x + b.x) * w.x + leaky02(a.y + b.y) * w.y +
               leaky02(a.z + b.z) * w.z + leaky02(a.w + b.w) * w.w;
    }
    esc[t] = acc;
    atomicMaxFloat(&m[d * NHEAD + h], acc);
}

// ---------------- edge pass 2: exp(e - max) + segment sum -----------------
__global__ __launch_bounds__(256) void k_edge_exp(
    const int* __restrict__ dst, float* __restrict__ esc,
    const float* __restrict__ m, float* __restrict__ denom, int EH)
{
    int t = blockIdx.x * blockDim.x + threadIdx.x;
    if (t >= EH) return;
    const int e = t >> 3, h = t & 7;
    const int d = dst[e];
    float ex = expf(esc[t] - m[d * NHEAD + h]);
    esc[t] = ex;
    atomicAdd(&denom[d * NHEAD + h], ex);
}

// ---------------- edge pass 3: weighted message scatter-add ---------------
__global__ __launch_bounds__(256) void k_edge_agg(
    const float* __restrict__ xl, const int* __restrict__ src,
    const int* __restrict__ dst, const float* __restrict__ esc,
    const float* __restrict__ denom, float* __restrict__ xbuf, int EH)
{
    int t = blockIdx.x * blockDim.x + threadIdx.x;
    if (t >= EH) return;
    const int e = t >> 3, h = t & 7;
    const int s = src[e], d = dst[e];
    const float w = esc[t] / denom[d * NHEAD + h];
    const float4* pl = (const float4*)(xl + s * HID + h * DH);
    float* po = xbuf + d * HID + h * DH;
    #pragma unroll
    for (int q = 0; q < 4; ++q) {
        float4 a = pl[q];
        atomicAdd(po + q * 4 + 0, w * a.x);
        atomicAdd(po + q * 4 + 1, w * a.y);
        atomicAdd(po + q * 4 + 2, w * a.z);
        atomicAdd(po + q * 4 + 3, w * a.w);
    }
}

// ---------------- bias + ELU (in place) -----------------------------------
__global__ void k_bias_elu(float* __restrict__ xbuf,
                           const float* __restrict__ b, int total) {
    int i = blockIdx.x * blockDim.x + threadIdx.x;
    if (i >= total) return;
    xbuf[i] = eluf(xbuf[i] + b[i & (HID - 1)]);
}

// ---------------- pool init ------------------------------------------------
__global__ void k_init_pool(float* __restrict__ sum_pool,
                            float* __restrict__ max_pool, int total) {
    int i = blockIdx.x * blockDim.x + threadIdx.x;
    if (i >= total) return;
    sum_pool[i] = 0.0f;
    max_pool[i] = -INFINITY;
}

// ---------------- per-node gated pooling (one wave per node) --------------
__global__ __launch_bounds__(256) void k_node_pool(
    const float* __restrict__ x, const int* __restrict__ batch,
    const float* __restrict__ read_w, const float* __restrict__ read_b,
    float* __restrict__ sum_pool, float* __restrict__ max_pool, int n)
{
    const int node = blockIdx.x * (blockDim.x >> 5) + (threadIdx.x >> 5);
    const int lane = threadIdx.x & 31;
    if (node >= n) return;
    float4 v = ((const float4*)(x + node * HID))[lane];
    float part = v.x * read_w[lane * 4 + 0] + v.y * read_w[lane * 4 + 1] +
                 v.z * read_w[lane * 4 + 2] + v.w * read_w[lane * 4 + 3];
    part = waveReduceSum(part);
    const float gate = 1.0f / (1.0f + expf(-(part + read_b[0])));
    const int b = batch[node];
    float* sp = sum_pool + b * HID + lane * 4;
    float* mp = max_pool + b * HID + lane * 4;
    atomicAdd(sp + 0, gate * v.x); atomicMaxFloat(mp + 0, v.x);
    atomicAdd(sp + 1, gate * v.y); atomicMaxFloat(mp + 1, v.y);
    atomicAdd(sp + 2, gate * v.z); atomicMaxFloat(mp + 2, v.z);
    atomicAdd(sp + 3, gate * v.w); atomicMaxFloat(mp + 3, v.w);
}

// ---------------- per-graph head (one wave per graph) ---------------------
__global__ __launch_bounds__(256) void k_head(
    const float* __restrict__ sum_pool, const float* __restrict__ max_pool,
    const float* __restrict__ temps, const float* __restrict__ en_w,
    const float* __restrict__ en_b, const float* __restrict__ w1,
    const float* __restrict__ b1, const float* __restrict__ w2,
    const float* __restrict__ b2, float* __restrict__ out, int B, int K)
{
    __shared__ float smol[8][2 * HID];
    __shared__ float sdist[8][104];
    const int w    = threadIdx.x >> 5;
    const int lane = threadIdx.x & 31;
    const int g    = blockIdx.x * 8 + w;
    const bool act = (g < B);

    if (act) {
        #pragma unroll
        for (int j = lane; j < HID; j += 32) {
            float mv = max_pool[g * HID + j];
            mv = (mv >= -3.0e38f && mv <= 3.0e38f) ? mv : 0.0f; // finite guard
            smol[w][j]       = sum_pool[g * HID + j];
            smol[w][HID + j] = mv;
        }
    }
    __syncthreads();

    // energies -> softmax(-E/T) across K (<=104), 4 k-values per lane
    float ek[4];
    float T = act ? temps[g] : 1.0f;
    float lmax = -INFINITY;
    #pragma unroll
    for (int c = 0; c < 4; ++c) {
        const int k = c * 32 + lane;
        float acc = -INFINITY;
        if (act && k < K) {
            acc = en_b[k];
            for (int i = 0; i < 2 * HID; ++i) acc += smol[w][i] * en_w[i * K + k];
            acc = -acc / T;
        }
        ek[c] = acc;
        lmax = fmaxf(lmax, acc);
    }
    lmax = waveReduceMax(lmax);
    float lsum = 0.0f;
    #pragma unroll
    for (int c = 0; c < 4; ++c) {
        const int k = c * 32 + lane;
        float dv = (act && k < K) ? expf(ek[c] - lmax) : 0.0f;
        ek[c] = dv;
        lsum += dv;
    }
    lsum = waveReduceSum(lsum);
    if (act) {
        #pragma unroll
        for (int c = 0; c < 4; ++c) {
            const int k = c * 32 + lane;
            if (k < K) sdist[w][k] = ek[c] / lsum;
        }
    }
    __syncthreads();

    // MLP: h = elu(dist @ w1 + b1) (K->64), out = h @ w2 + b2 (64->1)
    float part = 0.0f;
    if (act) {
        #pragma unroll
        for (int c = 0; c < 2; ++c) {
            const int j = c * 32 + lane;
            float acc = b1[j];
            for (int k = 0; k < K; ++k) acc += sdist[w][k] * w1[k * 64 + j];
            part += eluf(acc) * w2[j];
        }
    }
    part = waveReduceSum(part);
    if (act && lane == 0) out[g] = part + b2[0];
}

// ---------------- launcher -------------------------------------------------
extern "C" void kernel_launch(void* const* d_in, const int* in_sizes, int n_in,
                              void* d_out, int out_size, void* d_ws, size_t ws_size,
                              hipStream_t stream) {
    const float* x_in   = (const float*)d_in[0];
    const int*   eidx   = (const int*)  d_in[1];
    const int*   batch  = (const int*)  d_in[2];
    const float* temps  = (const float*)d_in[3];
    const float* wl     = (const float*)d_in[4];
    const float* wr     = (const float*)d_in[5];
    const float* att    = (const float*)d_in[6];
    const float* gat_b  = (const float*)d_in[7];
    const float* read_w = (const float*)d_in[8];
    const float* read_b = (const float*)d_in[9];
    const float* en_w   = (const float*)d_in[10];
    const float* en_b   = (const float*)d_in[11];
    const float* w1     = (const float*)d_in[12];
    const float* b1     = (const float*)d_in[13];
    const float* w2     = (const float*)d_in[14];
    const float* b2     = (const float*)d_in[15];
    float*       out    = (float*)d_out;

    const int N = in_sizes[0] / HID;
    const int E = in_sizes[1] / 2;
    const int B = in_sizes[3];
    const int L = in_sizes[4] / (HID * HID);
    const int K = in_sizes[11];
    const int EH = E * NHEAD;

    const int* src = eidx;
    const int* dst = eidx + E;

    // workspace carve-up (floats); xl+xr+xbuf (~150 MB) stay L2-resident
    float* ws    = (float*)d_ws;
    float* xl    = ws;               ws += (size_t)N * HID;
    float* xr    = ws;               ws += (size_t)N * HID;
    float* xbuf  = ws;               ws += (size_t)N * HID;
    float* esc   = ws;               ws += (size_t)E * NHEAD;
    float* mbuf  = ws;               ws += (size_t)N * NHEAD;
    float* denom = ws;               ws += (size_t)N * NHEAD;
    float* sump  = ws;               ws += (size_t)B * HID;
    float* maxp  = ws;               ws += (size_t)B * HID;
    (void)ws_size; (void)n_in; (void)out_size;

    const int tiles      = (N + 15) / 16;
    const int gemmBlocks = (tiles + 3) / 4;     // 4 tiles x 2 matrices per block
    const int edgeBlocks = (EH + 255) / 256;
    const int nhBlocks   = (N * HID + 255) / 256;
    const int poolBlocks = (B * HID + 255) / 256;
    const int nodeBlocks = (N + 7) / 8;
    const int headBlocks = (B + 7) / 8;
    const size_t gemmLds = (size_t)2 * HID * WT_STRIDE * sizeof(float); // ~135 KB

    for (int l = 0; l < L; ++l) {
        const float* xcur = (l == 0) ? x_in : xbuf;
        k_gemm_dual<<<gemmBlocks, 256, gemmLds, stream>>>(
            xcur, wl + (size_t)l * HID * HID, wr + (size_t)l * HID * HID,
            xl, xr, N);
        k_init_layer<<<nhBlocks, 256, 0, stream>>>(xbuf, mbuf, denom, N);
        k_edge_score<<<edgeBlocks, 256, 0, stream>>>(
            xl, xr, src, dst, att + (size_t)l * NHEAD * DH, esc, mbuf, EH);
        k_edge_exp<<<edgeBlocks, 256, 0, stream>>>(dst, esc, mbuf, denom, EH);
        k_edge_agg<<<edgeBlocks, 256, 0, stream>>>(xl, src, dst, esc, denom, xbuf, EH);
        k_bias_elu<<<nhBlocks, 256, 0, stream>>>(xbuf, gat_b + (size_t)l * HID, N * HID);
    }

    k_init_pool<<<poolBlocks, 256, 0, stream>>>(sump, maxp, B * HID);
    k_node_pool<<<nodeBlocks, 256, 0, stream>>>(xbuf, batch, read_w, read_b, sump, maxp, N);
    k_head<<<headBlocks, 256, 0, stream>>>(sump, maxp, temps, en_w, en_b,
                                           w1, b1, w2, b2, out, B, K);
}